// maxs_3813930959300
// MI455X (gfx1250) — compile-verified
//
#include <hip/hip_runtime.h>

// Row-wise max -> equality mask (int32 0/1), rows of 512 fp32.
// One wave32 per row; each lane owns 16 floats (4 x b128 NT loads in one
// clause), row max via 5 ds_swizzle_b32 xor-butterfly steps (no bounds-check
// VALU overhead, no LDS memory touched), compare from registers, 4 x b128 NT
// stores. Pure streaming: ~822 MB traffic -> ~35 us floor at 23.3 TB/s.

typedef float f32x4 __attribute__((ext_vector_type(4)));
typedef int   i32x4 __attribute__((ext_vector_type(4)));

#define ROW_D      512
#define WAVES_PB   8          // 256 threads / 32 lanes
#define VECS_LANE  4          // 4 * float4 * 32 lanes = 512 floats per row

// ds_swizzle_b32 group-of-32 mode: offset[15]=0, xor_mask=offset[14:10],
// or_mask=offset[9:5]=0, and_mask=offset[4:0]=0x1f  ->  SWAPX<k> patterns.
#define SWIZZLE_XOR(m, k)                                                     \
  (m) = fmaxf((m), __int_as_float(__builtin_amdgcn_ds_swizzle(                \
                       __float_as_int(m), ((k) << 10) | 0x1f)))

__global__ __launch_bounds__(256) void rowmax_mask_kernel(
    const float* __restrict__ in, int* __restrict__ out, int nrows) {
  const int lane = threadIdx.x & 31;
  const int row  = blockIdx.x * WAVES_PB + (threadIdx.x >> 5);
  if (row >= nrows) return;

  const f32x4* __restrict__ rin  = (const f32x4*)(in  + (size_t)row * ROW_D);
  i32x4*       __restrict__ rout = (i32x4*)      (out + (size_t)row * ROW_D);

  // Coalesced streaming loads: lane L, chunk j -> vector index L + 32*j.
  // Non-temporal: data is touched exactly once; don't rinse L2/WGP$.
  f32x4 v[VECS_LANE];
#pragma unroll
  for (int j = 0; j < VECS_LANE; ++j)
    v[j] = __builtin_nontemporal_load(&rin[lane + 32 * j]);

  // Per-lane max over the 16 registered elements.
  float m = v[0].x;
#pragma unroll
  for (int j = 0; j < VECS_LANE; ++j) {
    m = fmaxf(m, v[j].x);
    m = fmaxf(m, v[j].y);
    m = fmaxf(m, v[j].z);
    m = fmaxf(m, v[j].w);
  }

  // Wave32 cross-lane max: 5 one-instruction xor exchanges via ds_swizzle.
  SWIZZLE_XOR(m, 16);
  SWIZZLE_XOR(m, 8);
  SWIZZLE_XOR(m, 4);
  SWIZZLE_XOR(m, 2);
  SWIZZLE_XOR(m, 1);

  // Mask from registers (no re-read) and stream out as int32 0/1.
#pragma unroll
  for (int j = 0; j < VECS_LANE; ++j) {
    i32x4 o;
    o.x = (v[j].x == m) ? 1 : 0;
    o.y = (v[j].y == m) ? 1 : 0;
    o.z = (v[j].z == m) ? 1 : 0;
    o.w = (v[j].w == m) ? 1 : 0;
    __builtin_nontemporal_store(o, &rout[lane + 32 * j]);
  }
}

extern "C" void kernel_launch(void* const* d_in, const int* in_sizes, int n_in,
                              void* d_out, int out_size, void* d_ws, size_t ws_size,
                              hipStream_t stream) {
  (void)n_in; (void)d_ws; (void)ws_size; (void)out_size;
  const float* in = (const float*)d_in[0];
  int* out        = (int*)d_out;          // reference output dtype is int32
  const int nrows = in_sizes[0] / ROW_D;  // 200704

  const int blocks = (nrows + WAVES_PB - 1) / WAVES_PB;  // 25088
  rowmax_mask_kernel<<<blocks, 256, 0, stream>>>(in, out, nrows);
}